// RecurrentMambaCell_2937757630768
// MI455X (gfx1250) — compile-verified
//
#include <hip/hip_runtime.h>
#include <stdint.h>

typedef float v4f __attribute__((ext_vector_type(4)));
typedef uint32_t u32;
typedef u32 v4u __attribute__((ext_vector_type(4)));
typedef u32 v8u __attribute__((ext_vector_type(8)));

#define B_DIM   8192
#define D_DIM   2048
#define THREADS 256
#define TILE_D  (THREADS * 4)   // 1024 channels per block (d-tile)
#define ROWS    16               // batch rows per block -> grid = 2 x 512

// LDS layout in floats: [ W (TILE_D*4) | b | A | Bp | C | D_skip ] = 9216 floats = 36KB
#define OFF_W   0
#define OFF_B   (TILE_D * 4)          // 4096
#define OFF_A   (OFF_B + TILE_D)      // 5120
#define OFF_BP  (OFF_A + TILE_D)      // 6144
#define OFF_C   (OFF_BP + TILE_D)     // 7168
#define OFF_DS  (OFF_C + TILE_D)      // 8192
#define SMEM_FLOATS (OFF_DS + TILE_D) // 9216

// Issue one TDM DMA: contiguous nfloats f32 from global -> LDS byte address.
// D# per CDNA5 ISA ch.8: group0 = {count|flags, lds_addr, ga_lo, ga_hi|type=2},
// group1 = {data_size=4B, tensor_dim0=n, tensor_dim1=1, tile_dim0=n, tile_dim1=1,
//           dim0_stride=n}. VADDR2/3 omitted (NULL) => 2D descriptor form.
__device__ __forceinline__ void tdm_load_1d(u32 lds_byte_addr, const float* gsrc, u32 nfloats) {
  uint64_t ga = (uint64_t)(uintptr_t)gsrc;
  v4u g0;
  g0[0] = 1u;                                          // count=1, is_restore=0, gather off
  g0[1] = lds_byte_addr;                               // lds_addr
  g0[2] = (u32)ga;                                     // global_addr[31:0]
  g0[3] = ((u32)(ga >> 32) & 0x01FFFFFFu) | (2u << 30);// global_addr[56:32] | type=2
  v8u g1;
  g1[0] = (2u << 16);                                  // data_size=2 (4 bytes), no flags
  g1[1] = (nfloats & 0xFFFFu) << 16;                   // tensor_dim0[15:0] @ bits 63:48
  g1[2] = (nfloats >> 16) | (1u << 16);                // tensor_dim0[31:16] | tensor_dim1=1
  g1[3] = (nfloats & 0xFFFFu) << 16;                   // tile_dim0 @ bits 127:112
  g1[4] = 1u;                                          // tile_dim1=1, tile_dim2=0
  g1[5] = nfloats;                                     // tensor_dim0_stride[31:0]
  g1[6] = 0u;                                          // stride hi / dim1_stride lo
  g1[7] = 0u;
  asm volatile("tensor_load_to_lds %0, %1" :: "s"(g0), "s"(g1) : "memory");
}

__global__ __launch_bounds__(THREADS) void mamba_step_kernel(
    const float* __restrict__ x_t,        const float* __restrict__ conv_state,
    const float* __restrict__ ssm_state,  const float* __restrict__ W_conv,
    const float* __restrict__ b_conv,     const float* __restrict__ A,
    const float* __restrict__ Bp,         const float* __restrict__ C,
    const float* __restrict__ D_skip,
    float* __restrict__ y_out, float* __restrict__ conv_out_state,
    float* __restrict__ ssm_out)
{
  __shared__ __align__(16) float smem[SMEM_FLOATS];
  const u32 d0 = blockIdx.x * TILE_D;

  // --- Stage parameter tile to LDS via Tensor Data Mover (wave 0 issues) ---
  if (threadIdx.x == 0) {               // TDM ignores EXEC: issues once for the wave
    u32 lds0 = (u32)(uintptr_t)&smem[0];  // generic->LDS offset: low 32 bits
    tdm_load_1d(lds0 + OFF_W  * 4, W_conv + (size_t)d0 * 4, TILE_D * 4);
    tdm_load_1d(lds0 + OFF_B  * 4, b_conv + d0, TILE_D);
    tdm_load_1d(lds0 + OFF_A  * 4, A      + d0, TILE_D);
    tdm_load_1d(lds0 + OFF_BP * 4, Bp     + d0, TILE_D);
    tdm_load_1d(lds0 + OFF_C  * 4, C      + d0, TILE_D);
    tdm_load_1d(lds0 + OFF_DS * 4, D_skip + d0, TILE_D);
    __builtin_amdgcn_s_wait_tensorcnt(0); // scalar op: executes regardless of EXEC
  }
  __syncthreads();

  // --- Per-thread loop-invariant params from LDS (4 consecutive channels) ---
  const u32 c0 = threadIdx.x * 4;
  const v4f* sW4 = (const v4f*)&smem[OFF_W + c0 * 4];
  const v4f w0 = sW4[0], w1 = sW4[1], w2 = sW4[2], w3 = sW4[3];
  const v4f bb = *(const v4f*)&smem[OFF_B  + c0];
  const v4f aa = *(const v4f*)&smem[OFF_A  + c0];
  const v4f bp = *(const v4f*)&smem[OFF_BP + c0];
  const v4f cc = *(const v4f*)&smem[OFF_C  + c0];
  const v4f dd = *(const v4f*)&smem[OFF_DS + c0];

  const u32 ch = d0 + c0;

  for (int r = 0; r < ROWS; ++r) {
    const u32 b = blockIdx.y * ROWS + (u32)r;
    const size_t base  = (size_t)b * D_DIM + ch;   // x / ssm / y element index
    const size_t cbase = base * 3;                 // conv_state element index (K-1 = 3)

    if (r + 1 < ROWS) {                            // warm L2 for next row's conv stream
      __builtin_prefetch(conv_state + cbase + (size_t)D_DIM * 3, 0, 1);
    }

    // Streaming NT loads (single-use data: bypass cache retention)
    const v4f xv = __builtin_nontemporal_load((const v4f*)(x_t + base));
    const v4f sv = __builtin_nontemporal_load((const v4f*)(ssm_state + base));
    const v4f* cvp = (const v4f*)(conv_state + cbase);      // 48B-aligned
    const v4f cA = __builtin_nontemporal_load(cvp);         // cv0..cv3
    const v4f cB = __builtin_nontemporal_load(cvp + 1);     // cv4..cv7
    const v4f cC = __builtin_nontemporal_load(cvp + 2);     // cv8..cv11

    // conv_out[j] = dot(window_j, W_row_j) + b_j  (window_j = [cs0,cs1,cs2,x])
    v4f co;
    co.x = cA.x * w0.x + cA.y * w0.y + cA.z * w0.z + xv.x * w0.w + bb.x;
    co.y = cA.w * w1.x + cB.x * w1.y + cB.y * w1.z + xv.y * w1.w + bb.y;
    co.z = cB.z * w2.x + cB.w * w2.y + cC.x * w2.z + xv.z * w2.w + bb.z;
    co.w = cC.y * w3.x + cC.z * w3.y + cC.w * w3.z + xv.w * w3.w + bb.w;

    const v4f ns = aa * sv + bp * co;   // new ssm state
    const v4f yv = cc * ns + dd * xv;   // output

    __builtin_nontemporal_store(yv, (v4f*)(y_out  + base));
    __builtin_nontemporal_store(ns, (v4f*)(ssm_out + base));

    // new_conv_state[j] = [cs1, cs2, x]  -> shuffled into 3 B128 stores
    const v4f o0 = {cA.y, cA.z, xv.x, cB.x};
    const v4f o1 = {cB.y, xv.y, cB.w, cC.x};
    const v4f o2 = {xv.z, cC.z, cC.w, xv.w};
    v4f* ovp = (v4f*)(conv_out_state + cbase);
    __builtin_nontemporal_store(o0, ovp);
    __builtin_nontemporal_store(o1, ovp + 1);
    __builtin_nontemporal_store(o2, ovp + 2);
  }
}

extern "C" void kernel_launch(void* const* d_in, const int* in_sizes, int n_in,
                              void* d_out, int out_size, void* d_ws, size_t ws_size,
                              hipStream_t stream) {
  (void)in_sizes; (void)n_in; (void)out_size; (void)d_ws; (void)ws_size;
  const float* x_t        = (const float*)d_in[0];
  const float* conv_state = (const float*)d_in[1];
  const float* ssm_state  = (const float*)d_in[2];
  const float* W_conv     = (const float*)d_in[3];
  const float* b_conv     = (const float*)d_in[4];
  const float* A          = (const float*)d_in[5];
  const float* Bp         = (const float*)d_in[6];
  const float* C          = (const float*)d_in[7];
  const float* D_skip     = (const float*)d_in[8];

  float* y_out   = (float*)d_out;                            // (B, D)
  float* conv_ns = y_out + (size_t)B_DIM * D_DIM;            // (B, D, 3)
  float* ssm_ns  = conv_ns + (size_t)B_DIM * D_DIM * 3;      // (B, D)

  dim3 grid(D_DIM / TILE_D, B_DIM / ROWS);   // (2, 512)
  mamba_step_kernel<<<grid, THREADS, 0, stream>>>(
      x_t, conv_state, ssm_state, W_conv, b_conv, A, Bp, C, D_skip,
      y_out, conv_ns, ssm_ns);
}